// SpMiddleFHD_33998961115505
// MI455X (gfx1250) — compile-verified
//
#include <hip/hip_runtime.h>
#include <hip/hip_bf16.h>

typedef float v2f __attribute__((ext_vector_type(2)));
typedef float v4f __attribute__((ext_vector_type(4)));
typedef float v8f __attribute__((ext_vector_type(8)));

// ---------------------------------------------------------------------------
// keys: linear voxel key  ((b*D + z)*H + y)*W + x   (coords are sorted, so
// keys are sorted ascending -> binary search lookup)
// ---------------------------------------------------------------------------
__global__ void k_keys(const int* __restrict__ coords, int N, int D, int H, int W,
                       int* __restrict__ keys) {
  int i = blockIdx.x * blockDim.x + threadIdx.x;
  if (i < N) {
    const int* c = coords + i * 4;
    keys[i] = ((c[0] * D + c[1]) * H + c[2]) * W + c[3];
  }
}

static __device__ __forceinline__ int lower_bound_i32(const int* __restrict__ keys,
                                                      int n, int q) {
  int lo = 0, hi = n;
  while (lo < hi) {
    int mid = (lo + hi) >> 1;
    if (keys[mid] < q) lo = mid + 1; else hi = mid;
  }
  return lo;
}

// ---------------------------------------------------------------------------
// gather 27-neighborhood (im2col).  scale=1: submanifold conv (in==out level)
// scale=2: strided conv (input coords = 2*out + off).  Missing/out-of-range
// neighbors -> zeros.  G layout: (Nout, 27*Cin) row-major, K index = j*Cin+c.
// Cin is always a multiple of 4 and rows are 16B aligned -> float4 copies.
// ---------------------------------------------------------------------------
__global__ void k_gather27(const float* __restrict__ feats, const int* __restrict__ outc,
                           const int* __restrict__ inKeys, int Nout, int Nin, int Cin,
                           int D, int H, int W, int scale, float* __restrict__ G) {
  int idx = blockIdx.x * blockDim.x + threadIdx.x;
  if (idx >= Nout * 27) return;
  int n = idx / 27, j = idx - n * 27;
  int dz = j / 9 - 1, dy = (j / 3) % 3 - 1, dx = j % 3 - 1;
  const int* c = outc + n * 4;
  int b = c[0];
  int z = c[1] * scale + dz, y = c[2] * scale + dy, x = c[3] * scale + dx;
  bool inb = (z >= 0) & (z < D) & (y >= 0) & (y < H) & (x >= 0) & (x < W);
  v4f* __restrict__ g4 = (v4f*)(G + (size_t)idx * Cin);
  int C4 = Cin >> 2;
  if (inb) {
    int q = ((b * D + z) * H + y) * W + x;
    int pos = lower_bound_i32(inKeys, Nin, q);
    if (pos > Nin - 1) pos = Nin - 1;
    if (inKeys[pos] == q) {
      const v4f* __restrict__ f4 = (const v4f*)(feats + (size_t)pos * Cin);
      for (int cc = 0; cc < C4; ++cc) g4[cc] = f4[cc];
      return;
    }
  }
  v4f zero = {};
  for (int cc = 0; cc < C4; ++cc) g4[cc] = zero;
}

// ---------------------------------------------------------------------------
// f32 WMMA GEMM:  C[M, 16*NT] = A[M,K] * B   using V_WMMA_F32_16X16X4_F32.
// One wave computes a 16 x (16*NT) strip: A fragment loaded once per K-step
// (contiguous pair -> b64 load), NT back-to-back WMMAs on NT accumulators.
// Branch-free inner loop (BT/NT are compile-time).
// A frag (16x4 f32): VGPR r, lane<16 -> A[M=lane][K=r];  lane>=16 -> K=r+2.
// B frag (4x16 f32): VGPR r, lane<16 -> B[K=r][N=lane];  lane>=16 -> K=r+2.
// C (16x16 f32):     VGPR r, lane<16 -> C[M=r][N=lane];  lane>=16 -> M=r+8.
// BT=0: B row-major (K, Ncol) with leading dim ldb.
// BT=1: B stored transposed (Ncol, K) with leading dim ldb.
// ---------------------------------------------------------------------------
template <int BT, int NT>
__global__ void k_wmma_gemm(const float* __restrict__ A, const float* __restrict__ B,
                            float* __restrict__ C, int M, int K, int ldb) {
  constexpr int NCOL = 16 * NT;
  int lane = threadIdx.x;
  int half = lane >> 4;     // 0: lanes 0-15, 1: lanes 16-31
  int l16  = lane & 15;
  int row0 = blockIdx.x * 16;

  int arow = row0 + l16;
  bool av  = arow < M;
  int arc  = av ? arow : (M - 1);
  const float* ap = A + (size_t)arc * K + half * 2;

  v8f acc[NT];
#pragma unroll
  for (int t = 0; t < NT; ++t) acc[t] = (v8f){};

  for (int k = 0; k < K; k += 4) {
    v2f a = *(const v2f*)ap;
    ap += 4;
    if (!av) { a.x = 0.0f; a.y = 0.0f; }
#pragma unroll
    for (int t = 0; t < NT; ++t) {
      int bc = l16 + 16 * t;
      v2f b;
      if (BT) {
        b = *(const v2f*)(B + (size_t)bc * ldb + k + half * 2);
      } else {
        int kk = k + half * 2;
        b.x = B[(size_t)kk * ldb + bc];
        b.y = B[(size_t)(kk + 1) * ldb + bc];
      }
      acc[t] = __builtin_amdgcn_wmma_f32_16x16x4_f32(false, a, false, b,
                                                     (short)0, acc[t], false, false);
    }
  }

#pragma unroll
  for (int t = 0; t < NT; ++t) {
#pragma unroll
    for (int r = 0; r < 8; ++r) {
      int cr = row0 + r + half * 8;
      if (cr < M) C[(size_t)cr * NCOL + l16 + 16 * t] = acc[t][r];
    }
  }
}

// ---------------------------------------------------------------------------
// train-mode BatchNorm1d over active voxels, deterministic fixed-order
// reduction (single workgroup; C divides 256).  gb = (2,C): gamma, beta.
// Writes ss[0..C-1]=scale, ss[C..2C-1]=shift.
// ---------------------------------------------------------------------------
__global__ void k_bn_stats(const float* __restrict__ x, int N, int C,
                           const float* __restrict__ gb, float* __restrict__ ss) {
  __shared__ float s_sum[256];
  __shared__ float s_sq[256];
  int t = threadIdx.x;
  int c = t % C, s = t / C, S = 256 / C;
  float sum = 0.f, sq = 0.f;
  for (int r = s; r < N; r += S) {
    float v = x[(size_t)r * C + c];
    sum += v; sq += v * v;
  }
  s_sum[t] = sum; s_sq[t] = sq;
  __syncthreads();
  if (t < C) {
    float ts = 0.f, tq = 0.f;
    for (int s2 = 0; s2 < S; ++s2) { ts += s_sum[s2 * C + t]; tq += s_sq[s2 * C + t]; }
    float mean = ts / (float)N;
    float var  = tq / (float)N - mean * mean;
    float sc   = gb[t] * rsqrtf(var + 1e-3f);
    ss[t]      = sc;
    ss[C + t]  = gb[C + t] - mean * sc;
  }
}

__global__ void k_bn_apply(float* __restrict__ x, int N, int C,
                           const float* __restrict__ ss) {
  int i = blockIdx.x * blockDim.x + threadIdx.x;
  if (i < N * C) {
    int c = i % C;
    float v = x[i] * ss[c] + ss[C + c];
    x[i] = v > 0.f ? v : 0.f;
  }
}

// ---------------------------------------------------------------------------
// points_mean = (b, x, y, z) from coords0 batch + first 3 voxel features
// ---------------------------------------------------------------------------
__global__ void k_points_mean(const int* __restrict__ c0, const float* __restrict__ vf,
                              int N0, float* __restrict__ pm) {
  int n = blockIdx.x * blockDim.x + threadIdx.x;
  if (n < N0) {
    pm[n * 4 + 0] = (float)c0[n * 4 + 0];
    pm[n * 4 + 1] = vf[n * 4 + 0];
    pm[n * 4 + 2] = vf[n * 4 + 1];
    pm[n * 4 + 3] = vf[n * 4 + 2];
  }
}

// ---------------------------------------------------------------------------
// three_nn (batch-masked, brute force over known voxels of a level) +
// three_interpolate, writing into the concat buffer column slice.
// ---------------------------------------------------------------------------
__global__ void k_interp3(const int* __restrict__ c0, const float* __restrict__ vf, int N0,
                          const int* __restrict__ ck, const float* __restrict__ feats,
                          int Nk, int C, float vsx, float vsy, float vsz,
                          float* __restrict__ cat, int catLd, int colOff) {
  int n = blockIdx.x * blockDim.x + threadIdx.x;
  if (n >= N0) return;
  int   ub = c0[n * 4 + 0];
  float ux = vf[n * 4 + 0], uy = vf[n * 4 + 1], uz = vf[n * 4 + 2];
  float d0 = 1e30f, d1 = 1e30f, d2 = 1e30f;
  int   i0 = 0, i1 = 0, i2 = 0;
  for (int i = 0; i < Nk; ++i) {
    int kb = ck[i * 4 + 0];
    float px =   0.0f + ((float)ck[i * 4 + 3] + 0.5f) * vsx;
    float py = -40.0f + ((float)ck[i * 4 + 2] + 0.5f) * vsy;
    float pz =  -3.0f + ((float)ck[i * 4 + 1] + 0.5f) * vsz;
    float dx = ux - px, dy = uy - py, dz = uz - pz;
    float d = dx * dx + dy * dy + dz * dz + (kb != ub ? 1e10f : 0.0f);
    if (d < d0)      { d2 = d1; i2 = i1; d1 = d0; i1 = i0; d0 = d; i0 = i; }
    else if (d < d1) { d2 = d1; i2 = i1; d1 = d;  i1 = i; }
    else if (d < d2) { d2 = d;  i2 = i; }
  }
  float w0 = 1.0f / (d0 + 1e-8f), w1 = 1.0f / (d1 + 1e-8f), w2 = 1.0f / (d2 + 1e-8f);
  float wsum = w0 + w1 + w2;
  w0 /= wsum; w1 /= wsum; w2 /= wsum;
  float* row = cat + (size_t)n * catLd + colOff;
  const float* f0 = feats + (size_t)i0 * C;
  const float* f1 = feats + (size_t)i1 * C;
  const float* f2 = feats + (size_t)i2 * C;
  for (int c = 0; c < C; ++c) row[c] = w0 * f0[c] + w1 * f1[c] + w2 * f2[c];
}

// ---------------------------------------------------------------------------
// scatter extra-conv output into dense (B, 64*D3, H3, W3) with D3=6,H3=W3=100
// ---------------------------------------------------------------------------
__global__ void k_scatter(const float* __restrict__ f, const int* __restrict__ c3,
                          int N3, float* __restrict__ xd) {
  int idx = blockIdx.x * blockDim.x + threadIdx.x;
  if (idx >= N3 * 64) return;
  int n = idx >> 6, c = idx & 63;
  int b = c3[n * 4 + 0], d = c3[n * 4 + 1], h = c3[n * 4 + 2], w = c3[n * 4 + 3];
  xd[((size_t)(b * 384 + c * 6 + d) * 100 + h) * 100 + w] = f[(size_t)n * 64 + c];
}

// ---------------------------------------------------------------------------
// heads: cls (N0,1) and reg (N0,3) from pw (N0,64)
// ---------------------------------------------------------------------------
__global__ void k_heads(const float* __restrict__ pw, const float* __restrict__ cls_w,
                        const float* __restrict__ reg_w, int N0,
                        float* __restrict__ cls, float* __restrict__ reg) {
  int n = blockIdx.x * blockDim.x + threadIdx.x;
  if (n >= N0) return;
  const float* p = pw + (size_t)n * 64;
  float s = 0.f, r0 = 0.f, r1 = 0.f, r2 = 0.f;
  for (int o = 0; o < 64; ++o) {
    float v = p[o];
    s  += v * cls_w[o];
    r0 += v * reg_w[0 * 64 + o];
    r1 += v * reg_w[1 * 64 + o];
    r2 += v * reg_w[2 * 64 + o];
  }
  cls[n] = s;
  reg[n * 3 + 0] = r0; reg[n * 3 + 1] = r1; reg[n * 3 + 2] = r2;
}

// ---------------------------------------------------------------------------

extern "C" void kernel_launch(void* const* d_in, const int* in_sizes, int n_in,
                              void* d_out, int out_size, void* d_ws, size_t ws_size,
                              hipStream_t stream) {
  const float* vf = (const float*)d_in[0];
  const int* c0 = (const int*)d_in[1];
  const int* c1 = (const int*)d_in[2];
  const int* c2 = (const int*)d_in[3];
  const int* c3 = (const int*)d_in[4];
  const float* w00 = (const float*)d_in[5];  const float* b00 = (const float*)d_in[6];
  const float* w01 = (const float*)d_in[7];  const float* b01 = (const float*)d_in[8];
  const float* wd0 = (const float*)d_in[9];  const float* bd0 = (const float*)d_in[10];
  const float* w10 = (const float*)d_in[11]; const float* b10 = (const float*)d_in[12];
  const float* w11 = (const float*)d_in[13]; const float* b11 = (const float*)d_in[14];
  const float* wd1 = (const float*)d_in[15]; const float* bd1 = (const float*)d_in[16];
  const float* w20 = (const float*)d_in[17]; const float* b20 = (const float*)d_in[18];
  const float* w21 = (const float*)d_in[19]; const float* b21 = (const float*)d_in[20];
  const float* w22 = (const float*)d_in[21]; const float* b22 = (const float*)d_in[22];
  const float* wd2 = (const float*)d_in[23]; const float* bd2 = (const float*)d_in[24];
  const float* w30 = (const float*)d_in[25]; const float* b30 = (const float*)d_in[26];
  const float* w31 = (const float*)d_in[27]; const float* b31 = (const float*)d_in[28];
  const float* w32 = (const float*)d_in[29]; const float* b32 = (const float*)d_in[30];
  const float* we  = (const float*)d_in[31]; const float* be  = (const float*)d_in[32];
  const float* fcw = (const float*)d_in[33];
  const float* clw = (const float*)d_in[34];
  const float* rgw = (const float*)d_in[35];

  const int N0 = in_sizes[0] / 4;
  const int N1 = in_sizes[2] / 4;
  const int N2 = in_sizes[3] / 4;
  const int N3 = in_sizes[4] / 4;

  static const int DIM0[3] = {41, 800, 800};
  static const int DIM1[3] = {21, 400, 400};
  static const int DIM2[3] = {11, 200, 200};
  static const int DIM3[3] = {6, 100, 100};

  // ---- workspace layout ----
  char* base = (char*)d_ws;
  size_t off = 0;
  auto alloc = [&](size_t bytes) -> char* {
    char* p = base + off;
    off = (off + bytes + 255) & ~(size_t)255;
    return p;
  };
  size_t maxF = (size_t)N0 * 16;
  if ((size_t)N1 * 32 > maxF) maxF = (size_t)N1 * 32;
  if ((size_t)N2 * 64 > maxF) maxF = (size_t)N2 * 64;
  if ((size_t)N3 * 64 > maxF) maxF = (size_t)N3 * 64;
  size_t maxG = (size_t)N0 * 27 * 16;
  if ((size_t)N1 * 27 * 32 > maxG) maxG = (size_t)N1 * 27 * 32;
  if ((size_t)N2 * 27 * 64 > maxG) maxG = (size_t)N2 * 27 * 64;
  if ((size_t)N3 * 27 * 64 > maxG) maxG = (size_t)N3 * 27 * 64;

  int* k0 = (int*)alloc((size_t)N0 * 4);
  int* k1 = (int*)alloc((size_t)N1 * 4);
  int* k2 = (int*)alloc((size_t)N2 * 4);
  int* k3 = (int*)alloc((size_t)N3 * 4);
  float* fA  = (float*)alloc(maxF * 4);
  float* fB  = (float*)alloc(maxF * 4);
  float* G   = (float*)alloc(maxG * 4);
  float* cat = (float*)alloc((size_t)N0 * 160 * 4);
  float* pw  = (float*)alloc((size_t)N0 * 64 * 4);
  float* ss  = (float*)alloc(128 * 4);

  // ---- output layout: (xd | points_mean | cls | reg) ----
  const size_t XD = (size_t)2 * 384 * 100 * 100;   // 7,680,000
  float* out = (float*)d_out;
  float* pm  = out + XD;
  float* cls = pm + (size_t)N0 * 4;
  float* reg = cls + (size_t)N0;

  hipMemsetAsync(d_out, 0, XD * sizeof(float), stream);

  const int TB = 256;
  auto cdiv = [](int a, int b) { return (a + b - 1) / b; };

  k_keys<<<cdiv(N0, TB), TB, 0, stream>>>(c0, N0, DIM0[0], DIM0[1], DIM0[2], k0);
  k_keys<<<cdiv(N1, TB), TB, 0, stream>>>(c1, N1, DIM1[0], DIM1[1], DIM1[2], k1);
  k_keys<<<cdiv(N2, TB), TB, 0, stream>>>(c2, N2, DIM2[0], DIM2[1], DIM2[2], k2);
  k_keys<<<cdiv(N3, TB), TB, 0, stream>>>(c3, N3, DIM3[0], DIM3[1], DIM3[2], k3);
  k_points_mean<<<cdiv(N0, TB), TB, 0, stream>>>(c0, vf, N0, pm);

  auto gath = [&](const float* feats, const int* outc, const int* inKeys,
                  int Nout, int Nin, int Cin, const int* dims, int scale) {
    k_gather27<<<cdiv(Nout * 27, TB), TB, 0, stream>>>(
        feats, outc, inKeys, Nout, Nin, Cin, dims[0], dims[1], dims[2], scale, G);
  };
  auto gemm = [&](const float* A, const float* B, float* C, int M, int Ncol,
                  int K, int ldb, int bT) {
    dim3 grid(cdiv(M, 16));
    if (bT)             k_wmma_gemm<1, 4><<<grid, 32, 0, stream>>>(A, B, C, M, K, ldb);
    else if (Ncol == 16) k_wmma_gemm<0, 1><<<grid, 32, 0, stream>>>(A, B, C, M, K, ldb);
    else if (Ncol == 32) k_wmma_gemm<0, 2><<<grid, 32, 0, stream>>>(A, B, C, M, K, ldb);
    else                 k_wmma_gemm<0, 4><<<grid, 32, 0, stream>>>(A, B, C, M, K, ldb);
  };
  auto bn = [&](float* x, int N, int C, const float* gb) {
    k_bn_stats<<<1, 256, 0, stream>>>(x, N, C, gb, ss);
    k_bn_apply<<<cdiv(N * C, TB), TB, 0, stream>>>(x, N, C, ss);
  };

  // --- level 0: subm(4->16), subm(16->16) ---
  gath(vf, c0, k0, N0, N0, 4, DIM0, 1);
  gemm(G, w00, fA, N0, 16, 27 * 4, 16, 0);  bn(fA, N0, 16, b00);
  gath(fA, c0, k0, N0, N0, 16, DIM0, 1);
  gemm(G, w01, fB, N0, 16, 27 * 16, 16, 0); bn(fB, N0, 16, b01);

  // --- down 0->1 (16->32) ---
  gath(fB, c1, k0, N1, N0, 16, DIM0, 2);
  gemm(G, wd0, fA, N1, 32, 27 * 16, 32, 0); bn(fA, N1, 32, bd0);

  // --- level 1: subm x2 (32->32) ---
  gath(fA, c1, k1, N1, N1, 32, DIM1, 1);
  gemm(G, w10, fB, N1, 32, 27 * 32, 32, 0); bn(fB, N1, 32, b10);
  gath(fB, c1, k1, N1, N1, 32, DIM1, 1);
  gemm(G, w11, fA, N1, 32, 27 * 32, 32, 0); bn(fA, N1, 32, b11);
  // p1 -> cat cols [0,32)
  k_interp3<<<cdiv(N0, TB), TB, 0, stream>>>(c0, vf, N0, c1, fA, N1, 32,
                                             0.2f, 0.2f, 0.4f, cat, 160, 0);

  // --- down 1->2 (32->64) ---
  gath(fA, c2, k1, N2, N1, 32, DIM1, 2);
  gemm(G, wd1, fB, N2, 64, 27 * 32, 64, 0); bn(fB, N2, 64, bd1);

  // --- level 2: subm x3 (64->64) ---
  gath(fB, c2, k2, N2, N2, 64, DIM2, 1);
  gemm(G, w20, fA, N2, 64, 27 * 64, 64, 0); bn(fA, N2, 64, b20);
  gath(fA, c2, k2, N2, N2, 64, DIM2, 1);
  gemm(G, w21, fB, N2, 64, 27 * 64, 64, 0); bn(fB, N2, 64, b21);
  gath(fB, c2, k2, N2, N2, 64, DIM2, 1);
  gemm(G, w22, fA, N2, 64, 27 * 64, 64, 0); bn(fA, N2, 64, b22);
  // p2 -> cat cols [32,96)
  k_interp3<<<cdiv(N0, TB), TB, 0, stream>>>(c0, vf, N0, c2, fA, N2, 64,
                                             0.4f, 0.4f, 0.8f, cat, 160, 32);

  // --- down 2->3 (64->64) ---
  gath(fA, c3, k2, N3, N2, 64, DIM2, 2);
  gemm(G, wd2, fB, N3, 64, 27 * 64, 64, 0); bn(fB, N3, 64, bd2);

  // --- level 3: subm x3 (64->64) ---
  gath(fB, c3, k3, N3, N3, 64, DIM3, 1);
  gemm(G, w30, fA, N3, 64, 27 * 64, 64, 0); bn(fA, N3, 64, b30);
  gath(fA, c3, k3, N3, N3, 64, DIM3, 1);
  gemm(G, w31, fB, N3, 64, 27 * 64, 64, 0); bn(fB, N3, 64, b31);
  gath(fB, c3, k3, N3, N3, 64, DIM3, 1);
  gemm(G, w32, fA, N3, 64, 27 * 64, 64, 0); bn(fA, N3, 64, b32);
  // p3 -> cat cols [96,160)
  k_interp3<<<cdiv(N0, TB), TB, 0, stream>>>(c0, vf, N0, c3, fA, N3, 64,
                                             0.8f, 0.8f, 1.6f, cat, 160, 96);

  // --- extra 1x1x1 conv (64->64) + BN + scatter to dense ---
  gemm(fA, we, fB, N3, 64, 64, 64, 0); bn(fB, N3, 64, be);
  k_scatter<<<cdiv(N3 * 64, TB), TB, 0, stream>>>(fB, c3, N3, out);

  // --- point head: pw = cat(160) @ fc_w.T ; cls / reg ---
  gemm(cat, fcw, pw, N0, 64, 160, 160, 1);
  k_heads<<<cdiv(N0, TB), TB, 0, stream>>>(pw, clw, rgw, N0, cls, reg);
}